// GATConv_3917010174719
// MI455X (gfx1250) — compile-verified
//
#include <hip/hip_runtime.h>
#include <hip/hip_bf16.h>
#include <math.h>

typedef __attribute__((ext_vector_type(2))) float v2f;
typedef __attribute__((ext_vector_type(8))) float v8f;

#define N_NODES   100000
#define N_EDGES   1600000
#define IN_FEATS  128
#define ATTN_FEATS 64
#define NUM_HEADS 4
#define OUT_FEATS 32
#define HF        (NUM_HEADS * OUT_FEATS)   // 128
#define NEG_SLOPE 0.2f

// ---------------------------------------------------------------------------
// Kernel 1: fold attn_l / attn_r into the attention projection weight.
// wl[h][k] = sum_f fc_attn_w[(h*32+f)*64 + k] * attn_l[h*32+f]   (4x64 each)
// ---------------------------------------------------------------------------
__global__ __launch_bounds__(256) void gat_prep_attn(
    const float* __restrict__ fc_attn_w, const float* __restrict__ attn_l,
    const float* __restrict__ attn_r, float* __restrict__ wl,
    float* __restrict__ wr) {
  int t = threadIdx.x;              // 256 = 4 heads * 64 k
  int h = t >> 6;
  int k = t & 63;
  float al = 0.f, ar = 0.f;
  #pragma unroll 8
  for (int f = 0; f < OUT_FEATS; ++f) {
    float w = fc_attn_w[(h * OUT_FEATS + f) * ATTN_FEATS + k];
    al += w * attn_l[h * OUT_FEATS + f];
    ar += w * attn_r[h * OUT_FEATS + f];
  }
  wl[h * ATTN_FEATS + k] = al;
  wr[h * ATTN_FEATS + k] = ar;
}

// ---------------------------------------------------------------------------
// Kernel 2: feat_src = feat @ relu(W)^T via V_WMMA_F32_16X16X4_F32.
// Block = 8 waves; wave w computes the 16x16 tile (rows node0..+15,
// cols w*16..+15). K = 128 in steps of 4.
//
// A layout (16x4 f32): lane<16 -> row=lane, K=k0+{0,1}; lane>=16 -> row=lane-16,
// K=k0+{2,3}  => per-lane float2 at feat[(node0+lane&15)*128 + k0 + (lane>>4)*2].
// B layout (4x16 f32) mirrors A with W rows = output columns (B[k][n]=W[col0+n][k]).
// C layout: VGPR r -> row r (lanes 0-15) / row r+8 (lanes 16-31), col = lane&15.
// ---------------------------------------------------------------------------
__global__ __launch_bounds__(256) void gat_fc_wmma(
    const float* __restrict__ feat, const float* __restrict__ W,
    float* __restrict__ feat_src) {
  const int node0 = blockIdx.x * 16;
  const int wave  = threadIdx.x >> 5;
  const int lane  = threadIdx.x & 31;
  const int col0  = wave * 16;
  const int l15   = lane & 15;
  const int khalf = lane >> 4;

  const float* aptr = feat + (size_t)(node0 + l15) * IN_FEATS + khalf * 2;
  const float* bptr = W    + (size_t)(col0  + l15) * IN_FEATS + khalf * 2;

  v8f c = {};
  #pragma unroll 8
  for (int k = 0; k < IN_FEATS; k += 4) {
    v2f a = *(const v2f*)(aptr + k);
    v2f bw = *(const v2f*)(bptr + k);
    v2f b;
    b.x = fmaxf(bw.x, 0.f);   // ReparameterizedLinear: relu(raw weight)
    b.y = fmaxf(bw.y, 0.f);
    c = __builtin_amdgcn_wmma_f32_16x16x4_f32(
        /*neg_a=*/false, a, /*neg_b=*/false, b,
        /*c_mod=*/(short)0, c, /*reuse_a=*/false, /*reuse_b=*/false);
  }

  const int mbase = khalf * 8;
  #pragma unroll
  for (int r = 0; r < 8; ++r) {
    feat_src[(size_t)(node0 + mbase + r) * HF + col0 + l15] = c[r];
  }
}

// ---------------------------------------------------------------------------
// Kernel 3: el[n,h] = attn_feat[n,:] . wl[h,:] ; er likewise. One thread/node.
// ---------------------------------------------------------------------------
__global__ __launch_bounds__(256) void gat_el_er(
    const float* __restrict__ attn_feat, const float* __restrict__ wl,
    const float* __restrict__ wr, float* __restrict__ el,
    float* __restrict__ er) {
  int n = blockIdx.x * blockDim.x + threadIdx.x;
  if (n >= N_NODES) return;
  float accl[NUM_HEADS] = {0.f, 0.f, 0.f, 0.f};
  float accr[NUM_HEADS] = {0.f, 0.f, 0.f, 0.f};
  const float* row = attn_feat + (size_t)n * ATTN_FEATS;
  #pragma unroll 4
  for (int k = 0; k < ATTN_FEATS; ++k) {
    float x = row[k];
    #pragma unroll
    for (int h = 0; h < NUM_HEADS; ++h) {
      accl[h] = fmaf(x, wl[h * ATTN_FEATS + k], accl[h]);
      accr[h] = fmaf(x, wr[h * ATTN_FEATS + k], accr[h]);
    }
  }
  #pragma unroll
  for (int h = 0; h < NUM_HEADS; ++h) {
    el[n * NUM_HEADS + h] = accl[h];
    er[n * NUM_HEADS + h] = accr[h];
  }
}

// ---------------------------------------------------------------------------
// Kernel 4: per-call init (graph replays must be deterministic):
// out = 0, m = -FLT_MAX-ish, denom = 0.
// ---------------------------------------------------------------------------
__global__ __launch_bounds__(256) void gat_init(
    float* __restrict__ out, float* __restrict__ m,
    float* __restrict__ denom) {
  int idx = blockIdx.x * blockDim.x + threadIdx.x;
  if (idx < N_NODES * HF) out[idx] = 0.f;
  if (idx < N_NODES * NUM_HEADS) {
    m[idx] = -3.0e38f;
    denom[idx] = 0.f;
  }
}

__device__ __forceinline__ float leaky(float e) {
  return e > 0.f ? e : NEG_SLOPE * e;
}

// ---------------------------------------------------------------------------
// Kernel 5: segment max over edges (one thread per edge*head).
// ---------------------------------------------------------------------------
__global__ __launch_bounds__(256) void gat_edge_max(
    const int* __restrict__ src, const int* __restrict__ dst,
    const float* __restrict__ el, const float* __restrict__ er,
    float* __restrict__ m) {
  int t = blockIdx.x * blockDim.x + threadIdx.x;
  if (t >= N_EDGES * NUM_HEADS) return;
  int e = t >> 2, h = t & 3;
  int s = src[e], d = dst[e];
  float v = leaky(el[s * NUM_HEADS + h] + er[d * NUM_HEADS + h]);
  atomicMax(&m[d * NUM_HEADS + h], v);   // global_atomic_max_num_f32
}

// ---------------------------------------------------------------------------
// Kernel 6: segment exp-sum (one thread per edge*head).
// ---------------------------------------------------------------------------
__global__ __launch_bounds__(256) void gat_edge_sum(
    const int* __restrict__ src, const int* __restrict__ dst,
    const float* __restrict__ el, const float* __restrict__ er,
    const float* __restrict__ m, float* __restrict__ denom) {
  int t = blockIdx.x * blockDim.x + threadIdx.x;
  if (t >= N_EDGES * NUM_HEADS) return;
  int e = t >> 2, h = t & 3;
  int s = src[e], d = dst[e];
  float v = leaky(el[s * NUM_HEADS + h] + er[d * NUM_HEADS + h]);
  float ex = __expf(v - m[d * NUM_HEADS + h]);
  atomicAdd(&denom[d * NUM_HEADS + h], ex);
}

// ---------------------------------------------------------------------------
// Kernel 7: normalized scatter-aggregate. One wave per edge; lane covers
// hF = i*32 + lane over 4 iterations (head == i). feat_src and out are
// both L2-resident (51 MB each < 192 MB L2), so atomics land in L2.
// ---------------------------------------------------------------------------
__global__ __launch_bounds__(256) void gat_edge_agg(
    const int* __restrict__ src, const int* __restrict__ dst,
    const float* __restrict__ el, const float* __restrict__ er,
    const float* __restrict__ m, const float* __restrict__ denom,
    const float* __restrict__ feat_src, float* __restrict__ out) {
  int wave  = (blockIdx.x * blockDim.x + threadIdx.x) >> 5;
  int lane  = threadIdx.x & 31;
  int nwave = (gridDim.x * blockDim.x) >> 5;
  for (int e = wave; e < N_EDGES; e += nwave) {
    int s = src[e], d = dst[e];
    const float* fs = feat_src + (size_t)s * HF;
    float* od = out + (size_t)d * HF;
    #pragma unroll
    for (int h = 0; h < NUM_HEADS; ++h) {
      float v = leaky(el[s * NUM_HEADS + h] + er[d * NUM_HEADS + h]);
      float ex = __expf(v - m[d * NUM_HEADS + h]);
      float a = ex / denom[d * NUM_HEADS + h];
      int hF = h * 32 + lane;
      atomicAdd(&od[hF], fs[hF] * a);    // global_atomic_add_f32
    }
  }
}

// ---------------------------------------------------------------------------
extern "C" void kernel_launch(void* const* d_in, const int* in_sizes, int n_in,
                              void* d_out, int out_size, void* d_ws,
                              size_t ws_size, hipStream_t stream) {
  const float* feat      = (const float*)d_in[0];
  const float* attn_feat = (const float*)d_in[1];
  const float* fc_w      = (const float*)d_in[2];
  const float* fc_attn_w = (const float*)d_in[3];
  const float* attn_l    = (const float*)d_in[4];
  const float* attn_r    = (const float*)d_in[5];
  const int*   src       = (const int*)d_in[6];
  const int*   dst       = (const int*)d_in[7];
  float* out = (float*)d_out;

  // workspace layout (floats)
  float* ws       = (float*)d_ws;
  float* feat_src = ws;                                   // N*128
  float* el       = feat_src + (size_t)N_NODES * HF;      // N*4
  float* er       = el + (size_t)N_NODES * NUM_HEADS;     // N*4
  float* m        = er + (size_t)N_NODES * NUM_HEADS;     // N*4
  float* denom    = m  + (size_t)N_NODES * NUM_HEADS;     // N*4
  float* wl       = denom + (size_t)N_NODES * NUM_HEADS;  // 256
  float* wr       = wl + NUM_HEADS * ATTN_FEATS;          // 256

  gat_prep_attn<<<1, 256, 0, stream>>>(fc_attn_w, attn_l, attn_r, wl, wr);

  gat_fc_wmma<<<N_NODES / 16, 256, 0, stream>>>(feat, fc_w, feat_src);

  gat_el_er<<<(N_NODES + 255) / 256, 256, 0, stream>>>(attn_feat, wl, wr, el, er);

  gat_init<<<(N_NODES * HF + 255) / 256, 256, 0, stream>>>(out, m, denom);

  int et = N_EDGES * NUM_HEADS;
  gat_edge_max<<<(et + 255) / 256, 256, 0, stream>>>(src, dst, el, er, m);
  gat_edge_sum<<<(et + 255) / 256, 256, 0, stream>>>(src, dst, el, er, m, denom);

  // one wave per edge, 8 waves per block
  gat_edge_agg<<<N_EDGES / 8, 256, 0, stream>>>(src, dst, el, er, m, denom,
                                                feat_src, out);
}